// WarpTranspose_7224134992438
// MI455X (gfx1250) — compile-verified
//
#include <hip/hip_runtime.h>

// Problem shape (fixed by the reference): B=8, H=512, W=512, C=16, fp32.
#define B_ 8
#define H_ 512
#define W_ 512
#define C_ 16
#define NPIX (B_ * H_ * W_)

typedef float v4f __attribute__((ext_vector_type(4)));
typedef float v2f __attribute__((ext_vector_type(2)));

// Non-returning, relaxed, device-scope fp32 atomic add.
// Goal: lower to a single global_atomic_add_f32 (no pre-op return, resolves at L2).
__device__ __forceinline__ void atomic_add_f32(float* p, float v) {
    __hip_atomic_fetch_add(p, v, __ATOMIC_RELAXED, __HIP_MEMORY_SCOPE_AGENT);
}

__global__ __launch_bounds__(256) void warp_adjoint_scatter(
    const float* __restrict__ grad,   // [B,H,W,C]
    const float* __restrict__ u,      // [B,H,W,2]
    float* __restrict__ out)          // [B,H,W,C], pre-zeroed
{
    int p = blockIdx.x * 256 + threadIdx.x;          // pixel index in [0, B*H*W)
    if (p >= NPIX) return;

    // Decode (b, y, x); W = H = 512 -> pure shifts/masks.
    int x = p & (W_ - 1);
    int y = (p >> 9) & (H_ - 1);
    size_t imgbase = (size_t)(p >> 18) * ((size_t)H_ * W_ * C_);   // b * H*W*C

    // Stream the 16 channels (read-once) with non-temporal hints so the
    // 128MB grad_in stream does not evict the output accumulators from L2.
    const v4f* g4 = (const v4f*)(grad + (size_t)p * C_);
    v4f g0 = __builtin_nontemporal_load(g4 + 0);
    v4f g1 = __builtin_nontemporal_load(g4 + 1);
    v4f g2 = __builtin_nontemporal_load(g4 + 2);
    v4f g3 = __builtin_nontemporal_load(g4 + 3);

    v2f uv = __builtin_nontemporal_load((const v2f*)(u + (size_t)p * 2));

    float px = (float)x + uv.x;
    float py = (float)y + uv.y;
    float x0f = floorf(px);
    float y0f = floorf(py);
    float dx = px - x0f;
    float dy = py - y0f;
    int x0 = (int)x0f;
    int y0 = (int)y0f;

    float w00 = (1.0f - dx) * (1.0f - dy);
    float w01 = dx * (1.0f - dy);
    float w10 = (1.0f - dx) * dy;
    float w11 = dx * dy;

    // Scatter-add w * g[0:16] onto one corner if in-bounds (mode='zeros').
    // (cy*W + cx)*C == ((cy<<9) | cx) << 4  for W=512, C=16.
#define SPLAT(cx, cy, wgt)                                                     \
    do {                                                                       \
        if ((unsigned)(cx) < (unsigned)W_ && (unsigned)(cy) < (unsigned)H_) {  \
            float* dst = out + imgbase +                                       \
                         ((((size_t)(unsigned)(cy) << 9) | (unsigned)(cx)) << 4);\
            atomic_add_f32(dst + 0,  (wgt) * g0.x);                            \
            atomic_add_f32(dst + 1,  (wgt) * g0.y);                            \
            atomic_add_f32(dst + 2,  (wgt) * g0.z);                            \
            atomic_add_f32(dst + 3,  (wgt) * g0.w);                            \
            atomic_add_f32(dst + 4,  (wgt) * g1.x);                            \
            atomic_add_f32(dst + 5,  (wgt) * g1.y);                            \
            atomic_add_f32(dst + 6,  (wgt) * g1.z);                            \
            atomic_add_f32(dst + 7,  (wgt) * g1.w);                            \
            atomic_add_f32(dst + 8,  (wgt) * g2.x);                            \
            atomic_add_f32(dst + 9,  (wgt) * g2.y);                            \
            atomic_add_f32(dst + 10, (wgt) * g2.z);                            \
            atomic_add_f32(dst + 11, (wgt) * g2.w);                            \
            atomic_add_f32(dst + 12, (wgt) * g3.x);                            \
            atomic_add_f32(dst + 13, (wgt) * g3.y);                            \
            atomic_add_f32(dst + 14, (wgt) * g3.z);                            \
            atomic_add_f32(dst + 15, (wgt) * g3.w);                            \
        }                                                                      \
    } while (0)

    SPLAT(x0,     y0,     w00);
    SPLAT(x0 + 1, y0,     w01);
    SPLAT(x0,     y0 + 1, w10);
    SPLAT(x0 + 1, y0 + 1, w11);
#undef SPLAT
}

extern "C" void kernel_launch(void* const* d_in, const int* in_sizes, int n_in,
                              void* d_out, int out_size, void* d_ws, size_t ws_size,
                              hipStream_t stream) {
    (void)n_in; (void)d_ws; (void)ws_size; (void)in_sizes;
    const float* grad = (const float*)d_in[0];   // [B,H,W,C] f32
    const float* u    = (const float*)d_in[1];   // [B,H,W,2] f32
    // d_in[2] (x) only supplies shape/dtype in the reference; unused here.
    float* out = (float*)d_out;                  // [B,H,W,C] f32

    // Output is an accumulator: must start from zeros every call.
    hipMemsetAsync(out, 0, (size_t)out_size * sizeof(float), stream);

    warp_adjoint_scatter<<<NPIX / 256, 256, 0, stream>>>(grad, u, out);
}